// RWKV_LAYER_53790170415693
// MI455X (gfx1250) — compile-verified
//
#include <hip/hip_runtime.h>
#include <hip/hip_bf16.h>

#define D 2048
#define H 8192
#define NLAYER 12
#define KLEN 1024          // K-slice per block
#define NW 8               // waves per gemv block
#define WK (KLEN / NW)     // K per wave = 128

typedef __attribute__((ext_vector_type(2))) float v2f;
typedef __attribute__((ext_vector_type(8))) float v8f;

__device__ __forceinline__ float sigmoidf_(float x) { return 1.0f / (1.0f + expf(-x)); }

// ---------------------------------------------------------------------------
// Core: one 256-thread block computes 16 output rows of y = W[rows x K] @ x
// for K-slice [kb, kb+KLEN), using V_WMMA_F32_16X16X4_F32.
//   A = 16x4 tile of W  (lanes 0-15: M, K={0,1}; lanes 16-31: M, K={2,3})
//   B = 4x16 broadcast of x (B[k][n] = x[k] for every column n)
//   C accumulates 16x16; column 0 (lane 0 rows 0-7, lane 16 rows 8-15) is y.
// 8 waves split the K-slice; LDS reduce; 16 partial sums written to out16.
// ---------------------------------------------------------------------------
__device__ __forceinline__ void wmma_gemv_core(const float* __restrict__ W, int K,
                                               int r0, int kb,
                                               const float* __restrict__ xs,
                                               float (*red)[16],
                                               float* __restrict__ out16) {
  const int tid  = threadIdx.x;
  const int wave = tid >> 5;
  const int lane = tid & 31;
  const int half = lane >> 4;   // which K pair this lane-half owns
  const int mrow = lane & 15;
  const int kw   = wave * WK;
  const float* wp = W + (size_t)(r0 + mrow) * (size_t)K + (size_t)(kb + kw + 2 * half);
  const float* xp = xs + kw + 2 * half;
  v8f c = {0.f, 0.f, 0.f, 0.f, 0.f, 0.f, 0.f, 0.f};
#pragma unroll 8
  for (int kk = 0; kk < WK; kk += 4) {
    v2f a = __builtin_nontemporal_load((const v2f*)(wp + kk));  // streamed weights
    v2f b = *(const v2f*)(xp + kk);                             // LDS-resident vector
    c = __builtin_amdgcn_wmma_f32_16x16x4_f32(false, a, false, b, (short)0, c,
                                              false, false);
  }
  // lanes 0 and 16 hold column N=0: rows r0..r0+7 and r0+8..r0+15
  if (mrow == 0) {
#pragma unroll
    for (int i = 0; i < 8; ++i) red[wave][half * 8 + i] = c[i];
  }
  __syncthreads();
  if (tid < 16) {
    float s = 0.f;
#pragma unroll
    for (int w = 0; w < NW; ++w) s += red[w][tid];
    out16[tid] = s;
  }
}

// ---------------------------------------------------------------------------
// LayerNorm + time-mix (attention branch). One block of 1024 threads, D=2048.
// ---------------------------------------------------------------------------
__global__ __launch_bounds__(1024) void ln1_mix_kernel(
    const float* __restrict__ xc, const float* __restrict__ s1,
    const float* __restrict__ w, const float* __restrict__ b,
    const float* __restrict__ tmk, const float* __restrict__ tmv,
    const float* __restrict__ tmr,
    float* __restrict__ xn_out, float* __restrict__ xk,
    float* __restrict__ xv, float* __restrict__ xr) {
  __shared__ float sred[1024];
  const int t = threadIdx.x;
  float x0 = xc[t], x1 = xc[t + 1024];
  sred[t] = x0 + x1;
  __syncthreads();
  for (int s = 512; s > 0; s >>= 1) { if (t < s) sred[t] += sred[t + s]; __syncthreads(); }
  float mu = sred[0] * (1.0f / 2048.0f);
  __syncthreads();
  float d0 = x0 - mu, d1 = x1 - mu;
  sred[t] = d0 * d0 + d1 * d1;
  __syncthreads();
  for (int s = 512; s > 0; s >>= 1) { if (t < s) sred[t] += sred[t + s]; __syncthreads(); }
  float rstd = rsqrtf(sred[0] * (1.0f / 2048.0f) + 1e-5f);
#pragma unroll
  for (int j = 0; j < 2; ++j) {
    int i = t + j * 1024;
    float dv = (j == 0) ? d0 : d1;
    float xn = dv * rstd * w[i] + b[i];
    float sv = s1[i];
    xn_out[i] = xn;
    xk[i] = xn + sv * tmk[i];
    xv[i] = xn + sv * tmv[i];
    xr[i] = xn + sv * tmr[i];
  }
}

// ---------------------------------------------------------------------------
// sx = xc + (output-gemv partial reduce); LayerNorm2 + FFN time-mix.
// ---------------------------------------------------------------------------
__global__ __launch_bounds__(1024) void ln2_mix_kernel(
    const float* __restrict__ xc, const float* __restrict__ po,
    const float* __restrict__ s0,
    const float* __restrict__ w, const float* __restrict__ b,
    const float* __restrict__ ftmk, const float* __restrict__ ftmr,
    float* __restrict__ xn2_out, float* __restrict__ sx_out,
    float* __restrict__ xk2, float* __restrict__ xr2) {
  __shared__ float sred[1024];
  const int t = threadIdx.x;
  float sx0 = xc[t] + po[t] + po[D + t];
  float sx1 = xc[t + 1024] + po[t + 1024] + po[D + t + 1024];
  sx_out[t] = sx0;
  sx_out[t + 1024] = sx1;
  sred[t] = sx0 + sx1;
  __syncthreads();
  for (int s = 512; s > 0; s >>= 1) { if (t < s) sred[t] += sred[t + s]; __syncthreads(); }
  float mu = sred[0] * (1.0f / 2048.0f);
  __syncthreads();
  float d0 = sx0 - mu, d1 = sx1 - mu;
  sred[t] = d0 * d0 + d1 * d1;
  __syncthreads();
  for (int s = 512; s > 0; s >>= 1) { if (t < s) sred[t] += sred[t + s]; __syncthreads(); }
  float rstd = rsqrtf(sred[0] * (1.0f / 2048.0f) + 1e-5f);
#pragma unroll
  for (int j = 0; j < 2; ++j) {
    int i = t + j * 1024;
    float dv = (j == 0) ? d0 : d1;
    float xn = dv * rstd * w[i] + b[i];
    float sv = s0[i];
    xn2_out[i] = xn;
    xk2[i] = xn + sv * ftmk[i];
    xr2[i] = xn + sv * ftmr[i];
  }
}

// ---------------------------------------------------------------------------
// WKV recurrence (elementwise over D). Reduces split-K partials of k/v/r.
// ---------------------------------------------------------------------------
__global__ __launch_bounds__(256) void wkv_kernel(
    const float* __restrict__ pk, const float* __restrict__ pv,
    const float* __restrict__ pr,
    const float* __restrict__ tf, const float* __restrict__ td,
    const float* __restrict__ st_in_l, float* __restrict__ st_out_l,
    float* __restrict__ rab) {
  int i = blockIdx.x * 256 + threadIdx.x;
  float k = pk[i] + pk[D + i];
  float v = pv[i] + pv[D + i];
  float r = sigmoidf_(pr[i] + pr[D + i]);
  float aa = st_in_l[2 * D + i];
  float bb = st_in_l[3 * D + i];
  float pp = st_in_l[4 * D + i];
  float ww = tf[i] + k;
  float p = fmaxf(pp, ww);
  float e1 = expf(pp - p), e2 = expf(ww - p);
  float a = e1 * aa + e2 * v;
  float bden = e1 * bb + e2;
  rab[i] = r * (a / bden);
  float ww2 = pp + td[i];
  float p2 = fmaxf(ww2, k);
  float e1b = expf(ww2 - p2), e2b = expf(k - p2);
  st_out_l[2 * D + i] = e1b * aa + e2b * v;
  st_out_l[3 * D + i] = e1b * bb + e2b;
  st_out_l[4 * D + i] = p2;
}

// ---------------------------------------------------------------------------
// Generic single-matrix GEMV (split-K partials). grid = (rows/16)*(K/KLEN).
// ---------------------------------------------------------------------------
__global__ __launch_bounds__(256) void gemv_one_kernel(
    const float* __restrict__ W, const float* __restrict__ x,
    float* __restrict__ part, int K, int rows) {
  __shared__ float xs[KLEN];
  __shared__ float red[NW][16];
  int ks = K / KLEN;
  int rg = blockIdx.x / ks, kg = blockIdx.x % ks;
  int r0 = rg * 16, kb = kg * KLEN;
  for (int j = threadIdx.x; j < KLEN; j += 256) xs[j] = x[kb + j];
  __syncthreads();
  wmma_gemv_core(W, K, r0, kb, xs, red, part + (size_t)kg * rows + r0);
}

// k/v/r GEMVs fused into one grid (3 x 128 row-groups x 2 K-slices = 768).
__global__ __launch_bounds__(256) void gemv3_kernel(
    const float* __restrict__ Wk, const float* __restrict__ Wv,
    const float* __restrict__ Wr,
    const float* __restrict__ xk, const float* __restrict__ xv,
    const float* __restrict__ xr,
    float* __restrict__ pk, float* __restrict__ pv, float* __restrict__ pr) {
  __shared__ float xs[KLEN];
  __shared__ float red[NW][16];
  int mat = blockIdx.x >> 8;
  int sub = blockIdx.x & 255;
  const float* W = (mat == 0) ? Wk : (mat == 1) ? Wv : Wr;
  const float* x = (mat == 0) ? xk : (mat == 1) ? xv : xr;
  float* part    = (mat == 0) ? pk : (mat == 1) ? pv : pr;
  int rg = sub >> 1, kg = sub & 1;
  int r0 = rg * 16, kb = kg * KLEN;
  for (int j = threadIdx.x; j < KLEN; j += 256) xs[j] = x[kb + j];
  __syncthreads();
  wmma_gemv_core(W, D, r0, kb, xs, red, part + (size_t)kg * D + r0);
}

// key_ffn (8192 rows) + receptance_ffn (2048 rows) fused: grid 1024 + 256.
__global__ __launch_bounds__(256) void gemv_ffn1_kernel(
    const float* __restrict__ Wfk, const float* __restrict__ Wfr,
    const float* __restrict__ xk2, const float* __restrict__ xr2,
    float* __restrict__ pfk, float* __restrict__ pfr) {
  __shared__ float xs[KLEN];
  __shared__ float red[NW][16];
  const float* W;
  const float* x;
  float* part;
  int rows, sub;
  if (blockIdx.x < 1024) { W = Wfk; x = xk2; part = pfk; rows = H; sub = blockIdx.x; }
  else                   { W = Wfr; x = xr2; part = pfr; rows = D; sub = blockIdx.x - 1024; }
  int rg = sub >> 1, kg = sub & 1;
  int r0 = rg * 16, kb = kg * KLEN;
  for (int j = threadIdx.x; j < KLEN; j += 256) xs[j] = x[kb + j];
  __syncthreads();
  wmma_gemv_core(W, D, r0, kb, xs, red, part + (size_t)kg * rows + r0);
}

// value_ffn GEMV with relu(fk)^2 fused into the LDS staging. grid = 128*8.
__global__ __launch_bounds__(256) void gemv_ffn2_kernel(
    const float* __restrict__ Wfv, const float* __restrict__ pfk,
    float* __restrict__ pkv) {
  __shared__ float xs[KLEN];
  __shared__ float red[NW][16];
  int rg = blockIdx.x >> 3, kg = blockIdx.x & 7;
  int r0 = rg * 16, kb = kg * KLEN;
  for (int j = threadIdx.x; j < KLEN; j += 256) {
    float t = pfk[kb + j] + pfk[H + kb + j];  // reduce key_ffn split-K partials
    t = fmaxf(t, 0.f);
    xs[j] = t * t;                            // square(relu(fk))
  }
  __syncthreads();
  wmma_gemv_core(Wfv, H, r0, kb, xs, red, pkv + (size_t)kg * D + r0);
}

// xout = sx + sigmoid(fr) * kv   (reduces split-K partials of fr and kv)
__global__ __launch_bounds__(256) void ffn_out_kernel(
    const float* __restrict__ sx, const float* __restrict__ pfr,
    const float* __restrict__ pkv, float* __restrict__ out) {
  int i = blockIdx.x * 256 + threadIdx.x;
  float fr = sigmoidf_(pfr[i] + pfr[D + i]);
  float kv = 0.f;
#pragma unroll
  for (int g = 0; g < 8; ++g) kv += pkv[g * D + i];
  out[i] = sx[i] + fr * kv;
}

// ---------------------------------------------------------------------------
extern "C" void kernel_launch(void* const* d_in, const int* in_sizes, int n_in,
                              void* d_out, int out_size, void* d_ws, size_t ws_size,
                              hipStream_t stream) {
  (void)in_sizes; (void)n_in; (void)out_size; (void)ws_size;
  const float* x_in       = (const float*)d_in[0];
  const float* st_in      = (const float*)d_in[1];
  const float* ln1w       = (const float*)d_in[2];
  const float* ln1b       = (const float*)d_in[3];
  const float* ln2w       = (const float*)d_in[4];
  const float* ln2b       = (const float*)d_in[5];
  const float* td         = (const float*)d_in[6];
  const float* tf         = (const float*)d_in[7];
  const float* tmk        = (const float*)d_in[8];
  const float* tmv        = (const float*)d_in[9];
  const float* tmr        = (const float*)d_in[10];
  const float* key_w      = (const float*)d_in[11];
  const float* value_w    = (const float*)d_in[12];
  const float* recept_w   = (const float*)d_in[13];
  const float* output_w   = (const float*)d_in[14];
  const float* ftmk       = (const float*)d_in[15];
  const float* ftmr       = (const float*)d_in[16];
  const float* key_ffn    = (const float*)d_in[17];
  const float* recept_ffn = (const float*)d_in[18];
  const float* value_ffn  = (const float*)d_in[19];

  float* xout_final = (float*)d_out;        // [D]
  float* st_out     = (float*)d_out + D;    // [5L, D]

  // workspace layout (floats): 34*D total = 272 KB
  float* ws  = (float*)d_ws;
  float* xk  = ws + 0 * D;
  float* xv  = ws + 1 * D;
  float* xr  = ws + 2 * D;
  float* rab = ws + 3 * D;
  float* sx  = ws + 4 * D;
  float* xk2 = ws + 5 * D;
  float* xr2 = ws + 6 * D;
  float* xcb = ws + 7 * D;    // x carried between layers
  float* pk  = ws + 8 * D;    // 2*D split-K partials
  float* pv  = ws + 10 * D;
  float* pr  = ws + 12 * D;
  float* po  = ws + 14 * D;
  float* pfr = ws + 16 * D;   // 2*D
  float* pfk = ws + 18 * D;   // 2*H = 8*D
  float* pkv = ws + 26 * D;   // 8*D

  for (int l = 0; l < NLAYER; ++l) {
    const float* xc    = (l == 0) ? x_in : xcb;
    const float* sl_in = st_in + (size_t)5 * l * D;
    float* sl_out      = st_out + (size_t)5 * l * D;
    size_t oD = (size_t)l * D;

    ln1_mix_kernel<<<1, 1024, 0, stream>>>(xc, sl_in + D, ln1w + oD, ln1b + oD,
                                           tmk + oD, tmv + oD, tmr + oD,
                                           sl_out + D, xk, xv, xr);
    gemv3_kernel<<<768, 256, 0, stream>>>(key_w + (size_t)l * D * D,
                                          value_w + (size_t)l * D * D,
                                          recept_w + (size_t)l * D * D,
                                          xk, xv, xr, pk, pv, pr);
    wkv_kernel<<<D / 256, 256, 0, stream>>>(pk, pv, pr, tf + oD, td + oD,
                                            sl_in, sl_out, rab);
    gemv_one_kernel<<<256, 256, 0, stream>>>(output_w + (size_t)l * D * D, rab,
                                             po, D, D);
    ln2_mix_kernel<<<1, 1024, 0, stream>>>(xc, po, sl_in, ln2w + oD, ln2b + oD,
                                           ftmk + oD, ftmr + oD,
                                           sl_out, sx, xk2, xr2);
    gemv_ffn1_kernel<<<1280, 256, 0, stream>>>(key_ffn + (size_t)l * H * D,
                                               recept_ffn + (size_t)l * D * D,
                                               xk2, xr2, pfk, pfr);
    gemv_ffn2_kernel<<<1024, 256, 0, stream>>>(value_ffn + (size_t)l * D * H,
                                               pfk, pkv);
    float* xo = (l == NLAYER - 1) ? xout_final : xcb;
    ffn_out_kernel<<<D / 256, 256, 0, stream>>>(sx, pfr, pkv, xo);
  }
}